// LearnedRouter_33638183862714
// MI455X (gfx1250) — compile-verified
//
#include <hip/hip_runtime.h>

typedef __attribute__((ext_vector_type(2))) float v2f;
typedef __attribute__((ext_vector_type(4))) float v4f;
typedef __attribute__((ext_vector_type(8))) float v8f;
typedef __attribute__((ext_vector_type(4))) int   v4i;

#define HDIM 4096
#define NEXP 64
#define TOPK 8
#define MB   128            // tokens per workgroup
#define KC   64             // K-chunk staged in LDS
#define LDST (KC + 4)       // padded LDS row stride: 272B, 16B-aligned, bank-skewed
#define NCHUNK (HDIM / KC)

// ---- gfx1250 async global->LDS path (guarded; falls back to sync staging) ----
#ifdef __AMDGCN__
#if __has_builtin(__builtin_amdgcn_global_load_async_to_lds_b128)
#define USE_ASYNC 1
#endif
#if __has_builtin(__builtin_amdgcn_s_wait_asynccnt)
#define WAIT_ASYNC() __builtin_amdgcn_s_wait_asynccnt(0)
#else
#define WAIT_ASYNC() asm volatile("s_wait_asynccnt 0x0" ::: "memory")
#endif
#else
#define WAIT_ASYNC()
#endif

#ifdef USE_ASYNC
#define NBUF 2
// builtin signature (decoded from clang diagnostic):
//   void (...)(int4 __device__* gaddr, int4 __shared__* ldsaddr, imm offset, imm cpol)
typedef __attribute__((address_space(1))) v4i* as1_v4i_p;
typedef __attribute__((address_space(3))) v4i* as3_v4i_p;
#define ASYNC_CP_B128(gp, lp) \
    __builtin_amdgcn_global_load_async_to_lds_b128((as1_v4i_p)(gp), (as3_v4i_p)(lp), 0, 0)
#else
#define NBUF 1
#endif

// ---------------------------------------------------------------------------
// Kernel 1: logits = x @ W^T via V_WMMA_F32_16X16X4_F32, double-buffered
// async global->LDS staging. Each WG: 128 tokens x 64 experts (8 waves,
// 16 tokens x 4 accumulator tiles per wave).
// ---------------------------------------------------------------------------
__global__ __launch_bounds__(256) void router_gemm_f32wmma(
    const float* __restrict__ x, const float* __restrict__ W,
    float* __restrict__ logits)
{
    __shared__ float ldsX[NBUF][MB * LDST];   // per buf: 128 x 68 f = 34816 B
    __shared__ float ldsW[NBUF][NEXP * LDST]; // per buf:  64 x 68 f = 17408 B

    const int tid  = threadIdx.x;
    const int lane = tid & 31;
    const int wave = tid >> 5;          // 0..7
    const int l16  = lane & 15;
    const int half = lane >> 4;         // 0: K pair (0,1), 1: K pair (2,3)
    const int rowBase = blockIdx.x * MB;

    // cooperative staging coords: 16 threads per 64-float row, b128 each
    const int ldr = tid >> 4;           // 0..15
    const int ldc = (tid & 15) * 4;     // float column 0..60

    // fragment offsets within a buffer (floats)
    const int axO = (wave * 16 + l16) * LDST + half * 2;
    const int b0O = ( 0 + l16) * LDST + half * 2;
    const int b1O = (16 + l16) * LDST + half * 2;
    const int b2O = (32 + l16) * LDST + half * 2;
    const int b3O = (48 + l16) * LDST + half * 2;

    v8f acc0 = {}, acc1 = {}, acc2 = {}, acc3 = {};

    auto stage = [&](int buf, int k0) {
#ifdef USE_ASYNC
        #pragma unroll
        for (int i = 0; i < MB / 16; ++i) {
            const int r = ldr + i * 16;
            ASYNC_CP_B128(x + (size_t)(rowBase + r) * HDIM + k0 + ldc,
                          &ldsX[buf][r * LDST + ldc]);
        }
        #pragma unroll
        for (int i = 0; i < NEXP / 16; ++i) {
            const int r = ldr + i * 16;
            ASYNC_CP_B128(W + (size_t)r * HDIM + k0 + ldc,
                          &ldsW[buf][r * LDST + ldc]);
        }
#else
        #pragma unroll
        for (int i = 0; i < MB / 16; ++i) {
            const int r = ldr + i * 16;
            v4f v = *(const v4f*)(x + (size_t)(rowBase + r) * HDIM + k0 + ldc);
            *(v4f*)(&ldsX[buf][r * LDST + ldc]) = v;
        }
        #pragma unroll
        for (int i = 0; i < NEXP / 16; ++i) {
            const int r = ldr + i * 16;
            v4f v = *(const v4f*)(W + (size_t)r * HDIM + k0 + ldc);
            *(v4f*)(&ldsW[buf][r * LDST + ldc]) = v;
        }
#endif
    };

    auto compute = [&](int buf) {
        #pragma unroll
        for (int kk = 0; kk < KC; kk += 4) {
            v2f a   = *(const v2f*)&ldsX[buf][axO + kk];
            v2f vb0 = *(const v2f*)&ldsW[buf][b0O + kk];
            v2f vb1 = *(const v2f*)&ldsW[buf][b1O + kk];
            v2f vb2 = *(const v2f*)&ldsW[buf][b2O + kk];
            v2f vb3 = *(const v2f*)&ldsW[buf][b3O + kk];
            acc0 = __builtin_amdgcn_wmma_f32_16x16x4_f32(false, a, false, vb0,
                                                         (short)0, acc0, false, false);
            acc1 = __builtin_amdgcn_wmma_f32_16x16x4_f32(false, a, false, vb1,
                                                         (short)0, acc1, false, false);
            acc2 = __builtin_amdgcn_wmma_f32_16x16x4_f32(false, a, false, vb2,
                                                         (short)0, acc2, false, false);
            acc3 = __builtin_amdgcn_wmma_f32_16x16x4_f32(false, a, false, vb3,
                                                         (short)0, acc3, false, false);
        }
    };

#ifdef USE_ASYNC
    // prologue: fill buffer 0
    stage(0, 0);
    WAIT_ASYNC();
    __syncthreads();
    for (int c = 0; c < NCHUNK; ++c) {
        const int cur = c & 1;
        if (c + 1 < NCHUNK) stage(cur ^ 1, (c + 1) * KC);  // DMA overlaps WMMA
        compute(cur);
        WAIT_ASYNC();       // own async copies into 'nxt' complete
        __syncthreads();    // all waves done reading 'cur' / writing 'nxt'
    }
#else
    for (int c = 0; c < NCHUNK; ++c) {
        __syncthreads();
        stage(0, c * KC);
        __syncthreads();
        compute(0);
    }
#endif

    // C layout: VGPR r holds row (r + 8*half) for column N = l16 of each tile
    const int outRow = rowBase + wave * 16 + half * 8;
    #pragma unroll
    for (int r = 0; r < 8; ++r) {
        float* dst = logits + (size_t)(outRow + r) * NEXP;
        dst[ 0 + l16] = acc0[r];
        dst[16 + l16] = acc1[r];
        dst[32 + l16] = acc2[r];
        dst[48 + l16] = acc3[r];
    }
}

// ---------------------------------------------------------------------------
// Kernel 2: softmax over 64 experts + top-8 + L1 renorm. One wave32 / token.
// ---------------------------------------------------------------------------
__global__ __launch_bounds__(256) void router_softmax_topk(
    const float* __restrict__ logits, float* __restrict__ scores,
    float* __restrict__ weights, int* __restrict__ indices)
{
    const int token = blockIdx.x * 8 + (threadIdx.x >> 5);
    const int lane  = threadIdx.x & 31;

    const float* lr = logits + (size_t)token * NEXP;
    float v0 = lr[lane];
    float v1 = lr[lane + 32];

    float m = fmaxf(v0, v1);
    #pragma unroll
    for (int off = 16; off > 0; off >>= 1)
        m = fmaxf(m, __shfl_xor(m, off, 32));

    float e0 = __expf(v0 - m);
    float e1 = __expf(v1 - m);
    float s = e0 + e1;
    #pragma unroll
    for (int off = 16; off > 0; off >>= 1)
        s += __shfl_xor(s, off, 32);
    const float inv = __frcp_rn(s);

    float p0 = e0 * inv, p1 = e1 * inv;
    scores[(size_t)token * NEXP + lane]      = p0;
    scores[(size_t)token * NEXP + 32 + lane] = p1;

    // top-8 by iterative wave argmax with masking; ties -> lower index (jax)
    float c0 = p0, c1 = p1;
    float myw = 0.0f;
    int   myi = 0;
    float sum8 = 0.0f;
    #pragma unroll
    for (int k = 0; k < TOPK; ++k) {
        float val; int idx;
        if (c0 >= c1) { val = c0; idx = lane; }
        else          { val = c1; idx = lane + 32; }
        #pragma unroll
        for (int off = 16; off > 0; off >>= 1) {
            float ov = __shfl_xor(val, off, 32);
            int   oi = __shfl_xor(idx, off, 32);
            if (ov > val || (ov == val && oi < idx)) { val = ov; idx = oi; }
        }
        sum8 += val;
        if (lane == k) { myw = val; myi = idx; }
        if (idx == lane)      c0 = -1.0f;
        if (idx == lane + 32) c1 = -1.0f;
    }

    if (lane < TOPK) {
        weights[(size_t)token * TOPK + lane] = myw / sum8;
        indices[(size_t)token * TOPK + lane] = myi;
    }
}

// ---------------------------------------------------------------------------
extern "C" void kernel_launch(void* const* d_in, const int* in_sizes, int n_in,
                              void* d_out, int out_size, void* d_ws, size_t ws_size,
                              hipStream_t stream)
{
    const float* x = (const float*)d_in[0];   // [4,4096,4096] f32
    const float* W = (const float*)d_in[1];   // [64,4096] f32
    const int N = in_sizes[0] / HDIM;         // 16384 tokens

    float* out     = (float*)d_out;
    float* scores  = out;                                    // N*64
    float* logits  = out + (size_t)N * NEXP;                 // N*64
    float* weights = out + 2 * (size_t)N * NEXP;             // N*8
    int*   indices = (int*)(out + 2 * (size_t)N * NEXP + (size_t)N * TOPK); // N*8

    router_gemm_f32wmma<<<N / MB, 256, 0, stream>>>(x, W, logits);
    router_softmax_topk<<<N / 8, 256, 0, stream>>>(logits, scores, weights, indices);
}